// PaiNNInteraction_30588757082846
// MI455X (gfx1250) — compile-verified
//
#include <hip/hip_runtime.h>
#include <hip/hip_bf16.h>

typedef float v2f __attribute__((ext_vector_type(2)));
typedef float v8f __attribute__((ext_vector_type(8)));

#define N_ATOMS 50000
#define N_PAIRS 800000
#define FEAT    128
#define F3      384
#define LDSH    132   // padded row stride (floats) -> conflict-free ds_load_b64

// -------------------------------------------------------------------------
// Kernel 1: x = silu(q @ W1^T + b1) @ W2^T + b2   via V_WMMA_F32_16X16X4_F32
// One workgroup (256 thr = 8 waves) handles 16 atoms.
// A (16x4 f32): lane<16 holds row M=l, K = k0+{0,1}; lane>=16 holds K = k0+{2,3}
// B (4x16 f32): mirrored -> B[k][n] = W[n][k], contiguous float2 from weight row
// C/D (16x16 f32): VGPR r, lane<16 -> (M=r, N=l); lane>=16 -> (M=r+8, N=l-16)
// -------------------------------------------------------------------------
__global__ __launch_bounds__(256) void painn_dense(
    const float* __restrict__ q,
    const float* __restrict__ W1, const float* __restrict__ b1,
    const float* __restrict__ W2, const float* __restrict__ b2,
    float* __restrict__ x)
{
  __shared__ float hs[16 * LDSH];
  const int atom0 = blockIdx.x * 16;
  const int lane  = threadIdx.x & 31;
  const int wave  = threadIdx.x >> 5;   // 0..7
  const int half  = lane >> 4;          // 0 or 1
  const int l16   = lane & 15;

  // ---- stage 1: h = silu(q @ W1^T + b1); wave computes N-tile g0 = wave*16
  {
    const int g0 = wave * 16;
    v8f acc = {};
    const float* arow = q  + (size_t)(atom0 + l16) * FEAT + 2 * half;
    const float* brow = W1 + (size_t)(g0    + l16) * FEAT + 2 * half;
    #pragma unroll 8
    for (int k0 = 0; k0 < FEAT; k0 += 4) {
      v2f a = *(const v2f*)(arow + k0);
      v2f b = *(const v2f*)(brow + k0);
      acc = __builtin_amdgcn_wmma_f32_16x16x4_f32(false, a, false, b,
                                                  (short)0, acc, false, false);
    }
    const float bias = b1[g0 + l16];
    #pragma unroll
    for (int r = 0; r < 8; ++r) {
      float v = acc[r] + bias;
      // silu via hardware reciprocal (v_rcp_f32) instead of IEEE div sequence
      v = v * __builtin_amdgcn_rcpf(1.0f + __expf(-v));
      hs[(r + 8 * half) * LDSH + g0 + l16] = v;
    }
  }
  __syncthreads();

  // ---- stage 2: x = h @ W2^T + b2; wave handles N-tiles (wave*3+s)*16
  #pragma unroll
  for (int s = 0; s < 3; ++s) {
    const int g0 = (wave * 3 + s) * 16;
    v8f acc = {};
    const float* arow = hs + l16 * LDSH + 2 * half;
    const float* brow = W2 + (size_t)(g0 + l16) * FEAT + 2 * half;
    #pragma unroll 8
    for (int k0 = 0; k0 < FEAT; k0 += 4) {
      v2f a = *(const v2f*)(arow + k0);
      v2f b = *(const v2f*)(brow + k0);
      acc = __builtin_amdgcn_wmma_f32_16x16x4_f32(false, a, false, b,
                                                  (short)0, acc, false, false);
    }
    const float bias = b2[g0 + l16];
    #pragma unroll
    for (int r = 0; r < 8; ++r)
      x[(size_t)(atom0 + r + 8 * half) * F3 + g0 + l16] = acc[r] + bias;
  }
}

// -------------------------------------------------------------------------
// Kernel 2: initialize d_out with the identity terms: out_q = q, out_mu = mu
// -------------------------------------------------------------------------
__global__ __launch_bounds__(256) void painn_init(
    const float4* __restrict__ q4, const float4* __restrict__ mu4,
    float4* __restrict__ out4)
{
  const int nq4   = N_ATOMS * FEAT / 4;       // 1.6M
  const int ntot4 = N_ATOMS * 4 * FEAT / 4;   // 6.4M
  for (int i = blockIdx.x * blockDim.x + threadIdx.x; i < ntot4;
       i += gridDim.x * blockDim.x)
    out4[i] = (i < nq4) ? q4[i] : mu4[i - nq4];
}

// -------------------------------------------------------------------------
// Kernel 3: pair phase. 128 threads per pair (thread = feature), 2 pairs/block.
// Wij streamed non-temporally (1.23 GB once); x[j], mu[j], outputs stay in L2.
// -------------------------------------------------------------------------
__global__ __launch_bounds__(256) void painn_pairs(
    const float* __restrict__ x,    // [N, 384]
    const float* __restrict__ mu,   // [N, 3, 128]
    const float* __restrict__ Wij,  // [P, 384]
    const float* __restrict__ dir,  // [P, 3]
    const int*   __restrict__ pl,   // [2, P]
    float* __restrict__ outq,       // [N, 128]
    float* __restrict__ outmu)      // [N, 3, 128]
{
  const int p = blockIdx.x * 2 + (threadIdx.x >> 7);
  const int f = threadIdx.x & 127;
  if (p >= N_PAIRS) return;

  const int i = pl[p];
  const int j = pl[N_PAIRS + p];

  const float* w  = Wij + (size_t)p * F3;
  const float* xj = x   + (size_t)j * F3;
  const float* mj = mu  + (size_t)j * F3;

  const float wq = __builtin_nontemporal_load(w + f);
  const float wr = __builtin_nontemporal_load(w + FEAT + f);
  const float wm = __builtin_nontemporal_load(w + 2 * FEAT + f);

  const float dq    = wq * xj[f];
  const float dmuR  = wr * xj[FEAT + f];
  const float dmumu = wm * xj[2 * FEAT + f];

  const float d0 = dir[p * 3 + 0];
  const float d1 = dir[p * 3 + 1];
  const float d2 = dir[p * 3 + 2];

  atomicAdd(outq + (size_t)i * FEAT + f, dq);
  float* om = outmu + (size_t)i * F3;
  atomicAdd(om + f,            fmaf(dmuR, d0, dmumu * mj[f]));
  atomicAdd(om + FEAT + f,     fmaf(dmuR, d1, dmumu * mj[FEAT + f]));
  atomicAdd(om + 2 * FEAT + f, fmaf(dmuR, d2, dmumu * mj[2 * FEAT + f]));
}

// -------------------------------------------------------------------------
extern "C" void kernel_launch(void* const* d_in, const int* in_sizes, int n_in,
                              void* d_out, int out_size, void* d_ws, size_t ws_size,
                              hipStream_t stream)
{
  const float* q   = (const float*)d_in[0];
  const float* mu  = (const float*)d_in[1];
  const float* Wij = (const float*)d_in[2];
  const float* dir = (const float*)d_in[3];
  const int*   pl  = (const int*)  d_in[4];
  const float* W1  = (const float*)d_in[5];
  const float* b1  = (const float*)d_in[6];
  const float* W2  = (const float*)d_in[7];
  const float* b2  = (const float*)d_in[8];

  float* x     = (float*)d_ws;                       // 50000*384 f32 = 76.8 MB
  float* outq  = (float*)d_out;                      // [N,128]
  float* outmu = outq + (size_t)N_ATOMS * FEAT;      // [N,3,128]

  painn_dense<<<N_ATOMS / 16, 256, 0, stream>>>(q, W1, b1, W2, b2, x);
  painn_init <<<4096,        256, 0, stream>>>((const float4*)q, (const float4*)mu,
                                               (float4*)d_out);
  painn_pairs<<<N_PAIRS / 2, 256, 0, stream>>>(x, mu, Wij, dir, pl, outq, outmu);
}